// AdaptiveGraphicalModel_2654289788954
// MI455X (gfx1250) — compile-verified
//
#include <hip/hip_runtime.h>
#include <hip/hip_bf16.h>

// ---------------- problem constants ----------------
#define NJ   21
#define NE   40
#define BATCH 64
#define H    46
#define HH   (H*H)      // 2116
#define KS   45
#define KK   (KS*KS)    // 2025
#define PADK 22

// padded LDS image: rows 0..91 cover padded rows y0+r (y0<=32, r<=59)
#define INR  92
#define INCW 96
// kernel LDS: row w = ky+15, ky in [-15,59]; cols padded 45->48
#define KWR  75
#define KWC  48

typedef __attribute__((ext_vector_type(2))) float v2f;
typedef __attribute__((ext_vector_type(8))) float v8f;

// ---------------- tree tables ----------------
__constant__ int EDGE_FROM[NE] = {
  4,3,2,1,  8,7,6,5,  12,11,10,9, 16,15,14,13, 20,19,18,17,
  0,1,2,3,  0,5,6,7,  0,9,10,11,  0,13,14,15,  0,17,18,19 };
__constant__ int EDGE_NINC[NE] = {
  0,1,1,1, 0,1,1,1, 0,1,1,1, 0,1,1,1, 0,1,1,1,
  4,1,1,1, 4,1,1,1, 4,1,1,1, 4,1,1,1, 4,1,1,1 };
__constant__ int EDGE_INC[NE][4] = {
  {0,0,0,0},{0,0,0,0},{1,0,0,0},{2,0,0,0},
  {0,0,0,0},{4,0,0,0},{5,0,0,0},{6,0,0,0},
  {0,0,0,0},{8,0,0,0},{9,0,0,0},{10,0,0,0},
  {0,0,0,0},{12,0,0,0},{13,0,0,0},{14,0,0,0},
  {0,0,0,0},{16,0,0,0},{17,0,0,0},{18,0,0,0},
  {7,11,15,19},{20,0,0,0},{21,0,0,0},{22,0,0,0},
  {3,11,15,19},{24,0,0,0},{25,0,0,0},{26,0,0,0},
  {3,7,15,19}, {28,0,0,0},{29,0,0,0},{30,0,0,0},
  {3,7,11,19}, {32,0,0,0},{33,0,0,0},{34,0,0,0},
  {3,7,11,15}, {36,0,0,0},{37,0,0,0},{38,0,0,0} };
__constant__ int LEVELS[8][5] = {
  {0,4,8,12,16},{1,5,9,13,17},{2,6,10,14,18},{3,7,11,15,19},
  {20,24,28,32,36},{21,25,29,33,37},{22,26,30,34,38},{23,27,31,35,39} };
__constant__ int JINC_OFF[NJ+1] = {0,5,7,9,11,12,14,16,18,19,21,23,25,26,28,30,32,33,35,37,39,40};
__constant__ int JINC_LIST[NE] = {
  3,7,11,15,19, 2,20, 1,21, 0,22, 23, 6,24, 5,25, 4,26, 27,
  10,28, 9,29, 8,30, 31, 14,32, 13,33, 12,34, 35, 18,36, 17,37, 16,38, 39 };

// ---------------- per-map normalization factors ----------------
__global__ __launch_bounds__(256)
void bp_reduce_inv(const float* __restrict__ in, float* __restrict__ out,
                   int per, float floorv)
{
  __shared__ float red[256];
  const float* p = in + (size_t)blockIdx.x * per;
  float s = 0.f;
  for (int i = threadIdx.x; i < per; i += 256) s += fmaxf(p[i], floorv);
  red[threadIdx.x] = s; __syncthreads();
  for (int off = 128; off > 0; off >>= 1) {
    if (threadIdx.x < off) red[threadIdx.x] += red[threadIdx.x + off];
    __syncthreads();
  }
  if (threadIdx.x == 0) out[blockIdx.x] = 1.f / (red[0] + 1e-16f);
}

// ---------------- per-edge message: inter-belief -> conv(WMMA) -> normalize ----------------
__global__ __launch_bounds__(96)
void bp_edge_conv(const float* __restrict__ x, const float* __restrict__ kern,
                  const float* __restrict__ bias,
                  const float* __restrict__ invx, const float* __restrict__ invk,
                  float* __restrict__ msgs, int level)
{
  __shared__ float sIn[INR * INCW];   // 35328 B zero-padded inter-belief
  __shared__ float sKw[KWR * KWC];    // 14400 B zero-padded kernel (row = ky+15)
  __shared__ float sRed[96];

  const int slot = blockIdx.x % 5;
  const int b    = blockIdx.x / 5;
  const int e    = LEVELS[level][slot];
  const int fj   = EDGE_FROM[e];
  const int nin  = EDGE_NINC[e];
  const int tid  = threadIdx.x;

  const float* xp = x + ((size_t)b * NJ + fj) * HH;
  const float  xs = invx[b * NJ + fj];

  // stage 1a: inter-belief into padded LDS (log-domain value if nin==4)
  float lmax = -3.4e38f;
  for (int s = tid; s < INR * INCW; s += 96) {
    const int R = s / INCW, C = s % INCW;
    float v = 0.f;
    if (R >= PADK && R < PADK + H && C >= PADK && C < PADK + H) {
      const int p = (R - PADK) * H + (C - PADK);
      const float xv = fmaxf(xp[p], 1e-32f) * xs;
      if (nin == 0) {
        v = xv;
      } else if (nin == 1) {
        v = xv * msgs[((size_t)b * NE + EDGE_INC[e][0]) * HH + p];
      } else {
        float t = __logf(xv + 1e-32f);
        #pragma unroll
        for (int q = 0; q < 4; ++q)
          t += __logf(msgs[((size_t)b * NE + EDGE_INC[e][q]) * HH + p] + 1e-32f);
        v = t; lmax = fmaxf(lmax, t);
      }
    }
    sIn[s] = v;
  }

  // stage 1b: normalized kernel image into LDS
  const float* kp = kern + ((size_t)b * NE + e) * KK;
  const float  ks = invk[b * NE + e];
  for (int s = tid; s < KWR * KWC; s += 96) {
    const int r = s / KWC, c = s % KWC;
    const int ky = r - 15;
    float v = 0.f;
    if (ky >= 0 && ky < KS && c < KS) v = fmaxf(kp[ky * KS + c], 1e-16f) * ks;
    sKw[s] = v;
  }
  __syncthreads();

  if (nin == 4) {  // finish _mul_by_log_exp: exp(t - max)
    sRed[tid] = lmax; __syncthreads();
    if (tid == 0) {
      float m = -3.4e38f;
      for (int i = 0; i < 96; ++i) m = fmaxf(m, sRed[i]);
      sRed[0] = m;
    }
    __syncthreads();
    const float gmax = sRed[0];
    for (int s = tid; s < INR * INCW; s += 96) {
      const int R = s / INCW, C = s % INCW;
      if (R >= PADK && R < PADK + H && C >= PADK && C < PADK + H)
        sIn[s] = __expf(sIn[s] - gmax);
    }
    __syncthreads();
  }

  // stage 2: WMMA conv. wave w owns output rows [16w,16w+16), 3 tiles of 16 cols.
  // For input padded row y0+r: D[i][j] += sum_c A[i][c]*B[c][j],
  //   A[i][c] = w[r-i][k0+c]  (16 kernel rows x 4 taps)
  //   B[c][j] = in[y0+r][x0+j+k0+c]
  // Full unroll of the 12 K-chunks puts all 36 WMMAs of one input row in a
  // single straight-line block so the compiler can CSE the overlapping B
  // loads between column tiles (tile1@k0 == tile0@k0+16, tile2@k0 == tile0@k0+32).
  const int wave = tid >> 5;
  const int lane = tid & 31;
  const int n    = lane & 15;   // A: row m ; B/D: col n
  const int half = lane >> 4;   // selects tap pair {0,1}/{2,3}; D rows base 0/8
  const int y0   = wave * 16;

  v8f acc[3] = {v8f{0.f}, v8f{0.f}, v8f{0.f}};
  for (int r = 0; r < 60; ++r) {
    const float* kwRow = &sKw[(r - n + 15) * KWC + 2 * half];
    const float* inRow = &sIn[(y0 + r) * INCW + n + 2 * half];
    #pragma unroll
    for (int k0 = 0; k0 < 48; k0 += 4) {
      v2f a;  a.x  = kwRow[k0];      a.y  = kwRow[k0 + 1];
      v2f b0; b0.x = inRow[k0];      b0.y = inRow[k0 + 1];
      v2f b1; b1.x = inRow[k0 + 16]; b1.y = inRow[k0 + 17];
      v2f b2; b2.x = inRow[k0 + 32]; b2.y = inRow[k0 + 33];
      acc[0] = __builtin_amdgcn_wmma_f32_16x16x4_f32(false, a, false, b0,
                                                     (short)0, acc[0], false, false);
      acc[1] = __builtin_amdgcn_wmma_f32_16x16x4_f32(false, a, false, b1,
                                                     (short)0, acc[1], false, false);
      acc[2] = __builtin_amdgcn_wmma_f32_16x16x4_f32(false, a, false, b2,
                                                     (short)0, acc[2], false, false);
    }
  }

  // stage 3: normalize message: msg = (conv + bias) / (sum(conv + bias) + 1e-16)
  const float bval = bias[b * NE + e];
  float psum = 0.f;
  #pragma unroll
  for (int t = 0; t < 3; ++t) {
    const int X = t * 16 + n;
    #pragma unroll
    for (int v = 0; v < 8; ++v) {
      const int Y = y0 + half * 8 + v;
      if (Y < H && X < H) psum += acc[t][v];
    }
  }
  sRed[tid] = psum; __syncthreads();
  if (tid == 0) {
    float s = 0.f;
    for (int i = 0; i < 96; ++i) s += sRed[i];
    sRed[0] = 1.f / (s + (float)HH * bval + 1e-16f);
  }
  __syncthreads();
  const float invden = sRed[0];
  float* mp = msgs + ((size_t)b * NE + e) * HH;
  #pragma unroll
  for (int t = 0; t < 3; ++t) {
    const int X = t * 16 + n;
    if (X < H) {
      #pragma unroll
      for (int v = 0; v < 8; ++v) {
        const int Y = y0 + half * 8 + v;
        if (Y < H) mp[Y * H + X] = (acc[t][v] + bval) * invden;
      }
    }
  }
}

// ---------------- marginals ----------------
__global__ __launch_bounds__(256)
void bp_marginal(const float* __restrict__ x, const float* __restrict__ invx,
                 const float* __restrict__ msgs, float* __restrict__ out)
{
  __shared__ float t[HH];
  __shared__ float red[256];
  const int b = blockIdx.x / NJ, j = blockIdx.x % NJ;
  const float* xp = x + ((size_t)b * NJ + j) * HH;
  const float  xs = invx[b * NJ + j];
  const int o0 = JINC_OFF[j], o1 = JINC_OFF[j + 1];

  float lmax = -3.4e38f;
  for (int p = threadIdx.x; p < HH; p += 256) {
    float tv = __logf(fmaxf(xp[p], 1e-32f) * xs + 1e-32f);
    for (int q = o0; q < o1; ++q)
      tv += __logf(msgs[((size_t)b * NE + JINC_LIST[q]) * HH + p] + 1e-32f);
    t[p] = tv; lmax = fmaxf(lmax, tv);
  }
  red[threadIdx.x] = lmax; __syncthreads();
  for (int off = 128; off > 0; off >>= 1) {
    if (threadIdx.x < off) red[threadIdx.x] = fmaxf(red[threadIdx.x], red[threadIdx.x + off]);
    __syncthreads();
  }
  const float gmax = red[0]; __syncthreads();

  float s = 0.f;
  for (int p = threadIdx.x; p < HH; p += 256) {
    const float v = __expf(t[p] - gmax);
    t[p] = v; s += v;
  }
  red[threadIdx.x] = s; __syncthreads();
  for (int off = 128; off > 0; off >>= 1) {
    if (threadIdx.x < off) red[threadIdx.x] += red[threadIdx.x + off];
    __syncthreads();
  }
  const float invden = 1.f / (red[0] + 1e-16f);
  float* op = out + ((size_t)b * NJ + j) * HH;
  for (int p = threadIdx.x; p < HH; p += 256) op[p] = t[p] * invden;
}

// ---------------- launcher ----------------
extern "C" void kernel_launch(void* const* d_in, const int* in_sizes, int n_in,
                              void* d_out, int out_size, void* d_ws, size_t ws_size,
                              hipStream_t stream)
{
  const float* x    = (const float*)d_in[0];  // (64,21,46,46)
  const float* kern = (const float*)d_in[1];  // (64,40,45,45)
  const float* bias = (const float*)d_in[2];  // (64,40)
  float* out = (float*)d_out;                 // (64,21,46,46)

  float* msgs = (float*)d_ws;                          // 64*40*2116 floats
  float* invx = msgs + (size_t)BATCH * NE * HH;        // 64*21
  float* invk = invx + BATCH * NJ;                     // 64*40

  bp_reduce_inv<<<BATCH * NJ, 256, 0, stream>>>(x, invx, HH, 1e-32f);
  bp_reduce_inv<<<BATCH * NE, 256, 0, stream>>>(kern, invk, KK, 1e-16f);

  for (int lv = 0; lv < 8; ++lv)
    bp_edge_conv<<<BATCH * 5, 96, 0, stream>>>(x, kern, bias, invx, invk, msgs, lv);

  bp_marginal<<<BATCH * NJ, 256, 0, stream>>>(x, invx, msgs, out);
}